// NeighborGroup_45964740001822
// MI455X (gfx1250) — compile-verified
//
#include <hip/hip_runtime.h>
#include <hip/hip_bf16.h>
#include <math.h>

#define NPTS    8192
#define NQ      2048
#define NB      4
#define NCH     64
#define KNN     32
#define TQ      16      // queries per block (one WMMA M-tile)
#define CHUNK   512     // points per LDS distance tile
#define THREADS 512     // 16 waves: phase1 = 2 WMMA col-tiles/wave, phase2 = 1 query row/wave

typedef __attribute__((ext_vector_type(2))) float    v2f;
typedef __attribute__((ext_vector_type(8))) float    v8f;
typedef __attribute__((ext_vector_type(4))) uint32_t u32x4;
typedef __attribute__((ext_vector_type(8))) uint32_t u32x8;

// wave32 lane shuffle via LDS-permute hardware (no LDS memory touched)
static __device__ __forceinline__ float shflf(float x, int src) {
  return __int_as_float(__builtin_amdgcn_ds_bpermute(src << 2, __float_as_int(x)));
}
static __device__ __forceinline__ int shfli(int x, int src) {
  return __builtin_amdgcn_ds_bpermute(src << 2, x);
}
static __device__ __forceinline__ float readlane_f(float x, int l) {
  return __int_as_float(__builtin_amdgcn_readlane(__float_as_int(x), l));
}
static __device__ __forceinline__ float readfirst_f(float x) {
  return __int_as_float(__builtin_amdgcn_readfirstlane(__float_as_int(x)));
}

// one bitonic compare-exchange step on a lane-distributed (value,index) list
static __device__ __forceinline__ void cmpswap(float& v, int& i, int lane, int j, bool asc) {
  const int partner = lane ^ j;
  const float ov = shflf(v, partner);
  const int   oi = shfli(i, partner);
  const bool otherSmaller = (ov < v) || (ov == v && oi < i);  // tie-break: lower index first
  const bool iAmLow = (lane & j) == 0;
  const bool take = asc ? (iAmLow ? otherSmaller : !otherSmaller)
                        : (iAmLow ? !otherSmaller : otherSmaller);
  if (take) { v = ov; i = oi; }
}

// TDM: DMA one CHUNK of point coordinates (1536 f32 = 6 KB) from global to LDS.
// D# group0: count=1 | lds_addr | 57-bit global_addr | type=2("image").
// D# group1: data_size=4B, 1-D tile: tensor_dim0 = NPTS*3, tile_dim0 = CHUNK*3.
static __device__ __forceinline__ void tdm_load_chunk(const float* gsrc, uint32_t lds_byte_off) {
  const uint64_t ga = (uint64_t)(uintptr_t)gsrc;
  u32x4 g0;
  g0[0] = 1u;                                               // count=1, is_restore=0, no gather
  g0[1] = lds_byte_off;                                     // lds_addr
  g0[2] = (uint32_t)ga;                                     // global_addr[31:0]
  g0[3] = ((uint32_t)(ga >> 32) & 0x01FFFFFFu) | (2u << 30);// global_addr[56:32] | type=2
  u32x8 g1;
  g1[0] = (2u << 16);                                       // wg_mask=0, data_size=2 (4 bytes)
  g1[1] = (uint32_t)(NPTS * 3) << 16;                       // tensor_dim0[15:0] -> bits[31:16]
  g1[2] = ((uint32_t)(NPTS * 3) >> 16) | (1u << 16);        // tensor_dim0[31:16] | tensor_dim1=1
  g1[3] = (uint32_t)(CHUNK * 3) << 16;                      // tile_dim0 = 1536
  g1[4] = 0u;                                               // tile_dim1=0 (unused), tile_dim2=0
  g1[5] = (uint32_t)(NPTS * 3);                             // tensor_dim0_stride[31:0]
  g1[6] = 0u;
  g1[7] = 0u;
  asm volatile("tensor_load_to_lds %0, %1" :: "s"(g0), "s"(g1) : "memory");
}

__global__ __launch_bounds__(THREADS)
void knn_group_kernel(const float* __restrict__ xyz,
                      const float* __restrict__ new_xyz,
                      const float* __restrict__ feat,
                      float* __restrict__ out) {
  __shared__ float s_dist[TQ][CHUNK];    // 32 KB squared-distance tile
  __shared__ float s_pts[2][CHUNK * 3];  // 12 KB double-buffered TDM staging (xyz of chunk)
  __shared__ float s_q[TQ][4];           // qx,qy,qz,|q|^2

  const int tid  = threadIdx.x;
  const int lane = tid & 31;
  const int wave = tid >> 5;             // 0..15
  const int b    = blockIdx.x >> 7;      // 128 query-tiles per batch
  const int tile = blockIdx.x & 127;
  const int s0   = tile * TQ;

  const float* xb = xyz + (size_t)b * NPTS * 3;

  if (tid < TQ) {
    const float* q = new_xyz + ((size_t)b * NQ + s0 + tid) * 3;
    const float x = q[0], y = q[1], z = q[2];
    s_q[tid][0] = x; s_q[tid][1] = y; s_q[tid][2] = z;
    s_q[tid][3] = x * x + y * y + z * z;
  }
  // prefetch chunk 0 into buffer 0 via the Tensor Data Mover (wave-level op, EXEC ignored)
  if (wave == 0)
    tdm_load_chunk(xb, (uint32_t)(uintptr_t)&s_pts[0][0]);
  __syncthreads();

  const bool hi = lane >= 16;
  const int  m  = lane & 15;

  // A-matrix 16x4 f32: lanes 0-15 hold (K0,K1)=(qx,qy); lanes 16-31 hold (K2,K3)=(qz,0)
  v2f amat;
  amat.x = hi ? s_q[m][2] : s_q[m][0];
  amat.y = hi ? 0.0f      : s_q[m][1];

  float qn[8];
  #pragma unroll
  for (int i = 0; i < 8; ++i) qn[i] = s_q[i + (hi ? 8 : 0)][3];

  // running top-32, lane-distributed, sorted ascending by squared distance
  float topv   = __builtin_inff();
  int   topi   = 0;
  float curMax = __builtin_inff();       // scalar (readlane'd) kth-smallest bound

  int p = 0;
  for (int cb = 0; cb < NPTS; cb += CHUNK, p ^= 1) {
    // ---- TDM pipeline: kick chunk cb+CHUNK into the other buffer, wait for chunk cb ----
    if (wave == 0) {
      if (cb + CHUNK < NPTS) {
        tdm_load_chunk(xb + (size_t)(cb + CHUNK) * 3, (uint32_t)(uintptr_t)&s_pts[p ^ 1][0]);
        __builtin_amdgcn_s_wait_tensorcnt(1);   // previous DMA (current chunk) complete
      } else {
        __builtin_amdgcn_s_wait_tensorcnt(0);
      }
    }
    __syncthreads();   // publish s_pts[p] to all waves

    // ---- Phase 1: WMMA distance tiles into LDS (2 column-tiles per wave) ----
    #pragma unroll
    for (int t = 0; t < 2; ++t) {
      const int cloc = (wave * 2 + t) * 16 + m;           // column within chunk, owned by lane
      const float px = s_pts[p][cloc * 3 + 0];
      const float py = s_pts[p][cloc * 3 + 1];
      const float pz = s_pts[p][cloc * 3 + 2];
      const float pn = px * px + py * py + pz * pz;
      v2f bmat;                                           // B 4x16 f32, mirrors A by column
      bmat.x = hi ? pz   : px;
      bmat.y = hi ? 0.0f : py;
      v8f c = {0.f, 0.f, 0.f, 0.f, 0.f, 0.f, 0.f, 0.f};
      c = __builtin_amdgcn_wmma_f32_16x16x4_f32(false, amat, false, bmat,
                                                (short)0, c, false, false);
      #pragma unroll
      for (int i = 0; i < 8; ++i) {
        const int row = i + (hi ? 8 : 0);
        s_dist[row][cloc] = qn[i] + pn - 2.0f * c[i];     // |q-p|^2 (sqrt deferred)
      }
    }
    __syncthreads();

    // ---- Phase 2: wave r runs bitonic top-K for query row r ----
    {
      const float* drow = s_dist[wave];
      #pragma unroll 1
      for (int g = 0; g < CHUNK / 32; ++g) {
        float v   = drow[g * 32 + lane];
        int   idx = cb + g * 32 + lane;
        // wave-min prefilter, resolved as a scalar branch (s_cmp + s_cbranch)
        float vm = v;
        #pragma unroll
        for (int j = 16; j; j >>= 1) vm = fminf(vm, shflf(vm, lane ^ j));
        if (readfirst_f(vm) < curMax) {
          // full bitonic sort of 32 candidates, ascending
          #pragma unroll
          for (int k = 2; k <= 32; k <<= 1) {
            #pragma unroll
            for (int j = k >> 1; j; j >>= 1)
              cmpswap(v, idx, lane, j, (lane & k) == 0);
          }
          // reverse candidates; elementwise min with ascending top list -> bitonic
          const float rv = shflf(v, 31 - lane);
          const int   ri = shfli(idx, 31 - lane);
          if ((rv < topv) || (rv == topv && ri < topi)) { topv = rv; topi = ri; }
          // 5-stage bitonic merge restores ascending order
          #pragma unroll
          for (int j = 16; j; j >>= 1) cmpswap(topv, topi, lane, j, true);
          curMax = readlane_f(topv, 31);
        }
      }
    }
    __syncthreads();
  }

  // ---- Output: tuple concatenated flat (xyz, idx, feature, value) ----
  const size_t O_XYZ  = 0;
  const size_t O_IDX  = (size_t)NB * NQ * KNN * 3;             //   786432
  const size_t O_FEAT = O_IDX  + (size_t)NB * NQ * KNN;        //  1048576
  const size_t O_VAL  = O_FEAT + (size_t)NB * NQ * KNN * NCH;  // 17825792

  const int s = s0 + wave;
  const size_t base = ((size_t)b * NQ + s) * KNN + lane;       // lane = neighbor rank

  out[O_VAL + base] = sqrtf(fmaxf(topv, 0.0f));
  out[O_IDX + base] = (float)topi;

  const float* pt = xb + (size_t)topi * 3;
  float* oxyz = out + O_XYZ + base * 3;
  oxyz[0] = pt[0]; oxyz[1] = pt[1]; oxyz[2] = pt[2];

  const float* f  = feat + ((size_t)b * NPTS + topi) * NCH;    // L2-resident gather
  float*       of = out + O_FEAT + base * NCH;
  #pragma unroll
  for (int i = 0; i < NCH / 4; ++i)
    ((float4*)of)[i] = ((const float4*)f)[i];
}

extern "C" void kernel_launch(void* const* d_in, const int* in_sizes, int n_in,
                              void* d_out, int out_size, void* d_ws, size_t ws_size,
                              hipStream_t stream) {
  (void)in_sizes; (void)n_in; (void)d_ws; (void)ws_size; (void)out_size;
  const float* xyz     = (const float*)d_in[0];
  const float* new_xyz = (const float*)d_in[1];
  const float* feat    = (const float*)d_in[2];
  float* out = (float*)d_out;
  dim3 grid(NB * (NQ / TQ));   // 512 blocks
  dim3 block(THREADS);         // 16 waves
  hipLaunchKernelGGL(knn_group_kernel, grid, block, 0, stream, xyz, new_xyz, feat, out);
}